// topKsampling_4277787427259
// MI455X (gfx1250) — compile-verified
//
#include <hip/hip_runtime.h>
#include <stdint.h>

#define MUX_IN   8192
#define MUX_OUT  128
#define BATCH    64
#define EPS_F    1e-20f
#define T1       1024
#define T2       256

typedef float v4f __attribute__((ext_vector_type(4)));

// Types matching the gfx1250 async-DMA builtin prototype exactly:
//   void __builtin_amdgcn_global_load_async_to_lds_b128(
//        v4i addrspace(1)* gsrc, v4i addrspace(3)* ldst, imm offset, imm cpol)
typedef int v4i __attribute__((vector_size(16)));
typedef __attribute__((address_space(1))) v4i* as1_v4i_p;
typedef __attribute__((address_space(3))) v4i* as3_v4i_p;

// ---------------------------------------------------------------------------
// Kernel 1: one workgroup per batch row.
//  - async-DMA the shared logits vector into LDS (CDNA5 ASYNCcnt path)
//  - build order-preserving u64 keys: [monotone f32 bits | column index]
//  - full descending bitonic sort in LDS (8192 x u64 = 64 KB)
//  - emit the 128 leading (sorted) indices
// ---------------------------------------------------------------------------
__global__ __launch_bounds__(T1) void gumbel_topk_kernel(
    const float* __restrict__ logits,   // [MUX_IN]
    const float* __restrict__ u,        // [BATCH, MUX_IN]
    int* __restrict__ topk)             // [BATCH, MUX_OUT]
{
    __shared__ uint64_t keys[MUX_IN];   // 64 KB
    __shared__ float    slog[MUX_IN];   // 32 KB  (96 KB total < 320 KB WGP LDS)

    const int row = blockIdx.x;
    const int tid = (int)threadIdx.x;
    const float* __restrict__ urow = u + (size_t)row * MUX_IN;

    // gfx1250: global_prefetch_b8 for the u row this block is about to stream
    __builtin_prefetch(urow + tid * 8, 0, 1);

    // ---- stage logits into LDS via async global->LDS DMA ---------------------
#if __has_builtin(__builtin_amdgcn_global_load_async_to_lds_b128)
    for (int i = tid; i < MUX_IN / 4; i += T1) {
        __builtin_amdgcn_global_load_async_to_lds_b128(
            (as1_v4i_p)(uintptr_t)(logits + 4 * i),
            (as3_v4i_p)(uint32_t)(uintptr_t)(&slog[4 * i]),
            0, 0);
    }
#if __has_builtin(__builtin_amdgcn_s_wait_asynccnt)
    __builtin_amdgcn_s_wait_asynccnt(0);
#else
    asm volatile("s_wait_asynccnt 0" ::: "memory");
#endif
#else
    for (int i = tid; i < MUX_IN; i += T1) slog[i] = logits[i];
#endif
    __syncthreads();

    // ---- Gumbel perturbation + key pack --------------------------------------
    for (int i = tid; i < MUX_IN; i += T1) {
        float uu = urow[i];
        float gn = -__logf(-__logf(uu + EPS_F) + EPS_F);
        float v  = slog[i] + gn;
        uint32_t fb = __float_as_uint(v);
        // monotone map: float order -> unsigned order (no NaNs possible here)
        uint32_t m  = (uint32_t)(-(int32_t)(fb >> 31)) | 0x80000000u;
        keys[i] = ((uint64_t)(fb ^ m) << 32) | (uint32_t)i;
    }
    __syncthreads();

    // ---- bitonic sort, descending --------------------------------------------
    for (unsigned k = 2; k <= MUX_IN; k <<= 1) {
        for (unsigned j = k >> 1; j > 0; j >>= 1) {
#pragma unroll
            for (unsigned it = 0; it < (MUX_IN / 2) / T1; ++it) {
                unsigned t = (unsigned)tid + it * T1;
                unsigned i = 2u * t - (t & (j - 1u));
                unsigned p = i + j;
                uint64_t a = keys[i];
                uint64_t b = keys[p];
                // descending: keep max at i when (i & k) == 0
                if ((a < b) == ((i & k) == 0u)) { keys[i] = b; keys[p] = a; }
            }
            __syncthreads();
        }
    }

    if (tid < MUX_OUT)
        topk[row * MUX_OUT + tid] = (int)(uint32_t)(keys[tid] & 0xFFFFFFFFull);
}

// ---------------------------------------------------------------------------
// Kernel 2: one block per (row, k) slice of 8192 floats.
// Store-bandwidth bound (256 MiB) -> 16 B non-temporal stores, L2 bypass.
// `sel` is block-uniform -> scalar load.
// ---------------------------------------------------------------------------
__global__ __launch_bounds__(T2) void onehot_store_kernel(
    const int* __restrict__ topk,       // [BATCH * MUX_OUT]
    float* __restrict__ out)            // [BATCH, MUX_OUT, MUX_IN]
{
    const int slice = (int)blockIdx.x;          // row * MUX_OUT + k
    const int sel   = topk[slice];
    float* dst = out + (size_t)slice * MUX_IN;

#pragma unroll
    for (int it = 0; it < MUX_IN / (T2 * 4); ++it) {   // 8 iterations
        const int pos = (it * T2 + (int)threadIdx.x) * 4;
        v4f v;
        v.x = (pos + 0 == sel) ? 1.0f : 0.0f;
        v.y = (pos + 1 == sel) ? 1.0f : 0.0f;
        v.z = (pos + 2 == sel) ? 1.0f : 0.0f;
        v.w = (pos + 3 == sel) ? 1.0f : 0.0f;
        __builtin_nontemporal_store(v, (v4f*)(dst + pos));
    }
}

extern "C" void kernel_launch(void* const* d_in, const int* in_sizes, int n_in,
                              void* d_out, int out_size, void* d_ws, size_t ws_size,
                              hipStream_t stream) {
    const float* logits = (const float*)d_in[0];  // (1, 8192)
    // d_in[1] = inpData: unused by the reference computation
    const float* u      = (const float*)d_in[2];  // (64, 1, 8192)
    int* topk = (int*)d_ws;                       // 64*128*4 B = 32 KB scratch

    gumbel_topk_kernel<<<BATCH, T1, 0, stream>>>(logits, u, topk);
    onehot_store_kernel<<<BATCH * MUX_OUT, T2, 0, stream>>>(topk, (float*)d_out);

    (void)in_sizes; (void)n_in; (void)out_size; (void)ws_size;
}